// GAT_65317862637893
// MI455X (gfx1250) — compile-verified
//
#include <hip/hip_runtime.h>
#include <hip/hip_bf16.h>

typedef __attribute__((ext_vector_type(2))) float v2f;
typedef __attribute__((ext_vector_type(8))) float v8f;

#define HEADS 4
#define DIM 64
#define HD 256           // HEADS*DIM
#define NEG_SLOPE 0.2f

// ---- order-preserving float<->int key for atomic max over signed ints ----
__device__ __forceinline__ int fkey(float f) {
    int i = __float_as_int(f);
    return i >= 0 ? i : (i ^ 0x7FFFFFFF);
}
__device__ __forceinline__ float fdec(int k) {
    return k >= 0 ? __int_as_float(k) : __int_as_float(k ^ 0x7FFFFFFF);
}
#define NEG_INF_KEY 0x807FFFFF   // fkey(-inf)

// ============================================================
// 0) init emax (encoded -inf) and denom (0)
// ============================================================
__global__ void gat_init_kernel(int* __restrict__ emax, float* __restrict__ denom, int n4) {
    int i = blockIdx.x * blockDim.x + threadIdx.x;
    if (i < n4) { emax[i] = NEG_INF_KEY; denom[i] = 0.0f; }
}

// ============================================================
// 1) WMMA fp32 GEMM: feat = emb @ W_fc   [N,256]
//                    rst  = emb @ W_res + bias  (residual seed)
// one wave = 16 rows x 64 cols (4 accumulators), K=64 via 16 x V_WMMA_F32_16X16X4_F32
// wave id -> (rowTile, group); group 0..3 = W_fc col-quads, 4..7 = W_res col-quads
// ============================================================
__global__ void gat_gemm_wmma_kernel(const float* __restrict__ emb,
                                     const float* __restrict__ Wfc,
                                     const float* __restrict__ Wres,
                                     const float* __restrict__ bias,
                                     float* __restrict__ feat,
                                     float* __restrict__ rst,
                                     int nRowTiles) {
    const int lane   = threadIdx.x & 31;
    const int wave   = blockIdx.x * (blockDim.x >> 5) + (threadIdx.x >> 5);
    const int group  = wave & 7;
    const int rowTile = wave >> 3;
    if (rowTile >= nRowTiles) return;

    const int  m0      = rowTile * 16;
    const bool isRes   = (group >= 4);
    const int  c0      = (group & 3) * 64;
    const float* W     = isRes ? Wres : Wfc;
    float*     outp    = isRes ? rst  : feat;

    const int halfsel = lane >> 4;     // 0: lanes 0-15, 1: lanes 16-31
    const int l15     = lane & 15;

    // A fragment source: row = m0+l15, K pair = {k0+2*halfsel, k0+2*halfsel+1}
    const float* aBase = emb + (size_t)(m0 + l15) * DIM + 2 * halfsel;

    v8f acc0 = {}, acc1 = {}, acc2 = {}, acc3 = {};

    #pragma unroll
    for (int k0 = 0; k0 < DIM; k0 += 4) {
        v2f a;
        a.x = aBase[k0];
        a.y = aBase[k0 + 1];
        const int krow = k0 + 2 * halfsel;
        #pragma unroll
        for (int t = 0; t < 4; ++t) {
            const int col = c0 + t * 16 + l15;
            v2f b;
            b.x = W[(size_t)krow * HD + col];
            b.y = W[(size_t)(krow + 1) * HD + col];
            v8f c = (t == 0) ? acc0 : (t == 1) ? acc1 : (t == 2) ? acc2 : acc3;
            c = __builtin_amdgcn_wmma_f32_16x16x4_f32(false, a, false, b,
                                                      (short)0, c, false, false);
            if      (t == 0) acc0 = c;
            else if (t == 1) acc1 = c;
            else if (t == 2) acc2 = c;
            else             acc3 = c;
        }
    }

    // D layout: VGPR j -> row m0 + j + 8*halfsel, col = c0 + t*16 + l15
    #pragma unroll
    for (int t = 0; t < 4; ++t) {
        const int col = c0 + t * 16 + l15;
        const v8f acc = (t == 0) ? acc0 : (t == 1) ? acc1 : (t == 2) ? acc2 : acc3;
        const float bval = isRes ? bias[col] : 0.0f;
        #pragma unroll
        for (int j = 0; j < 8; ++j) {
            const int row = m0 + j + 8 * halfsel;
            outp[(size_t)row * HD + col] = acc[j] + bval;
        }
    }
}

// ============================================================
// 2) per-node attention logits: el[n,h] = feat[n,h,:]·attn_l[h,:], same for er
// one wave per node; lane covers 8 contiguous cols; 8-lane shfl reduce per head
// ============================================================
__global__ void gat_eler_kernel(const float* __restrict__ feat,
                                const float* __restrict__ attn_l,
                                const float* __restrict__ attn_r,
                                float* __restrict__ el,
                                float* __restrict__ er,
                                int nNodes) {
    const int lane = threadIdx.x & 31;
    const int node = blockIdx.x * (blockDim.x >> 5) + (threadIdx.x >> 5);
    if (node >= nNodes) return;

    const int cbase = lane * 8;          // flat col h*64+d
    const int h     = lane >> 3;
    const float* f  = feat + (size_t)node * HD + cbase;
    const float* al = attn_l + cbase;    // attn_l flat [H*D]
    const float* ar = attn_r + cbase;

    float sl = 0.0f, sr = 0.0f;
    #pragma unroll
    for (int i = 0; i < 8; ++i) {
        const float v = f[i];
        sl += v * al[i];
        sr += v * ar[i];
    }
    #pragma unroll
    for (int off = 1; off < 8; off <<= 1) {
        sl += __shfl_xor(sl, off, 32);
        sr += __shfl_xor(sr, off, 32);
    }
    if ((lane & 7) == 0) {
        el[node * HEADS + h] = sl;
        er[node * HEADS + h] = sr;
    }
}

// ============================================================
// 3) segment max: emax[dst,h] = max over edges of leaky(el[src]+er[dst])
// one thread per edge, int-encoded float atomicMax
// ============================================================
__global__ void gat_edge_max_kernel(const int* __restrict__ src,
                                    const int* __restrict__ dst,
                                    const float* __restrict__ el,
                                    const float* __restrict__ er,
                                    int* __restrict__ emax, int nE) {
    const int e = blockIdx.x * blockDim.x + threadIdx.x;
    if (e >= nE) return;
    const int s = src[e], d = dst[e];
    const float4 a = *(const float4*)(el + 4 * s);
    const float4 b = *(const float4*)(er + 4 * d);
    float v[4] = { a.x + b.x, a.y + b.y, a.z + b.z, a.w + b.w };
    #pragma unroll
    for (int h = 0; h < HEADS; ++h) {
        float x = v[h];
        x = x > 0.0f ? x : NEG_SLOPE * x;
        atomicMax(&emax[d * HEADS + h], fkey(x));
    }
}

// ============================================================
// 4) denom[dst,h] += exp(e - emax[dst,h])
// ============================================================
__global__ void gat_edge_expsum_kernel(const int* __restrict__ src,
                                       const int* __restrict__ dst,
                                       const float* __restrict__ el,
                                       const float* __restrict__ er,
                                       const int* __restrict__ emax,
                                       float* __restrict__ denom, int nE) {
    const int e = blockIdx.x * blockDim.x + threadIdx.x;
    if (e >= nE) return;
    const int s = src[e], d = dst[e];
    const float4 a = *(const float4*)(el + 4 * s);
    const float4 b = *(const float4*)(er + 4 * d);
    float v[4] = { a.x + b.x, a.y + b.y, a.z + b.z, a.w + b.w };
    #pragma unroll
    for (int h = 0; h < HEADS; ++h) {
        float x = v[h];
        x = x > 0.0f ? x : NEG_SLOPE * x;
        const float m  = fdec(emax[d * HEADS + h]);
        const float ex = __expf(x - m);
        unsafeAtomicAdd(&denom[d * HEADS + h], ex);
    }
}

// ============================================================
// 5) weighted scatter-aggregate: rst[dst] += alpha * feat[src]
// one wave per edge; lane handles 8 contiguous cols of its head
// ============================================================
__global__ void gat_edge_agg_kernel(const int* __restrict__ src,
                                    const int* __restrict__ dst,
                                    const float* __restrict__ el,
                                    const float* __restrict__ er,
                                    const int* __restrict__ emax,
                                    const float* __restrict__ denom,
                                    const float* __restrict__ feat,
                                    float* __restrict__ rst, int nE) {
    const int lane = threadIdx.x & 31;
    const int e    = blockIdx.x * (blockDim.x >> 5) + (threadIdx.x >> 5);
    if (e >= nE) return;
    const int s = src[e], d = dst[e];
    const int h = lane >> 3;

    float x = el[s * HEADS + h] + er[d * HEADS + h];
    x = x > 0.0f ? x : NEG_SLOPE * x;
    const float m     = fdec(emax[d * HEADS + h]);
    const float alpha = __expf(x - m) / denom[d * HEADS + h];

    const int cbase = lane * 8;
    const float* fs = feat + (size_t)s * HD + cbase;
    float*       rd = rst  + (size_t)d * HD + cbase;
    #pragma unroll
    for (int i = 0; i < 8; ++i) {
        unsafeAtomicAdd(rd + i, alpha * fs[i]);
    }
}

// ============================================================
// 6) out[n,d] = mean over heads of rst[n,h,d] (rst already has residual+bias)
// ============================================================
__global__ void gat_finalize_kernel(const float* __restrict__ rst,
                                    float* __restrict__ out, int total) {
    const int idx = blockIdx.x * blockDim.x + threadIdx.x;  // n*64 + d
    if (idx >= total) return;
    const int n = idx >> 6, dcol = idx & 63;
    const float* r = rst + (size_t)n * HD + dcol;
    out[idx] = 0.25f * (r[0] + r[64] + r[128] + r[192]);
}

// ============================================================
extern "C" void kernel_launch(void* const* d_in, const int* in_sizes, int n_in,
                              void* d_out, int out_size, void* d_ws, size_t ws_size,
                              hipStream_t stream) {
    const float* emb    = (const float*)d_in[0];
    const float* W_fc   = (const float*)d_in[1];
    const float* attn_l = (const float*)d_in[2];
    const float* attn_r = (const float*)d_in[3];
    const float* W_res  = (const float*)d_in[4];
    const float* bias   = (const float*)d_in[5];
    const int*   src    = (const int*)d_in[6];
    const int*   dst    = (const int*)d_in[7];

    const int nN = in_sizes[0] / DIM;   // 50000
    const int nE = in_sizes[6];         // 800000

    // workspace layout (fp32 elements)
    float* feat  = (float*)d_ws;                         // [N, 256]
    float* rst   = feat + (size_t)nN * HD;               // [N, 256]
    float* el    = rst  + (size_t)nN * HD;               // [N, 4]
    float* er    = el   + (size_t)nN * HEADS;            // [N, 4]
    int*   emax  = (int*)(er + (size_t)nN * HEADS);      // [N, 4] encoded
    float* denom = (float*)(emax + (size_t)nN * HEADS);  // [N, 4]

    // 0) init emax/denom
    {
        const int n4 = nN * HEADS;
        gat_init_kernel<<<(n4 + 255) / 256, 256, 0, stream>>>(emax, denom, n4);
    }
    // 1) WMMA GEMM: feat + residual-seeded rst
    {
        const int nRowTiles = (nN + 15) / 16;            // 3125
        const int nWaves    = nRowTiles * 8;             // 8 col-groups per row tile
        const int nBlocks   = (nWaves + 7) / 8;          // 8 waves / 256-thread block
        gat_gemm_wmma_kernel<<<nBlocks, 256, 0, stream>>>(emb, W_fc, W_res, bias,
                                                          feat, rst, nRowTiles);
    }
    // 2) el / er
    {
        const int nBlocks = (nN + 7) / 8;                // 8 nodes (waves) per block
        gat_eler_kernel<<<nBlocks, 256, 0, stream>>>(feat, attn_l, attn_r, el, er, nN);
    }
    // 3) segment max
    gat_edge_max_kernel<<<(nE + 255) / 256, 256, 0, stream>>>(src, dst, el, er, emax, nE);
    // 4) exp-sum denom
    gat_edge_expsum_kernel<<<(nE + 255) / 256, 256, 0, stream>>>(src, dst, el, er, emax, denom, nE);
    // 5) scatter-aggregate
    {
        const int nBlocks = (nE + 7) / 8;                // 8 edges (waves) per block
        gat_edge_agg_kernel<<<nBlocks, 256, 0, stream>>>(src, dst, el, er, emax, denom,
                                                         feat, rst, nE);
    }
    // 6) head-mean
    {
        const int total = nN * DIM;
        gat_finalize_kernel<<<(total + 255) / 256, 256, 0, stream>>>(rst, (float*)d_out, total);
    }
}